// NeuralHMM_33990371180551
// MI455X (gfx1250) — compile-verified
//
#include <hip/hip_runtime.h>
#include <hip/hip_bf16.h>
#include <math.h>

// ---------------------------------------------------------------------------
// NeuralHMM on MI455X (gfx1250): fp32 WMMA (V_WMMA_F32_16X16X4_F32) pipeline.
//   K=256 states, M=2, D=256, V=50000, T=512, B=256, Kp1=257.
// Workspace layout (floats), total ~120.9 MB:
//   decWT     @ 0           : 256 x 50000   (dec_W transposed)
//   tmWT      @ 12800000    : 512 x 512     (tm_W transposed)
//   tdWT      @ 13062144    : 512 x 256     (td_W transposed)
//   h_em      @ 13193216    : 256 x 256     (emission hidden, post-LN)
//   lsePart   @ 13258752    : 256 x 98 x 2  (per v-block max/sumexp partials)
//   lse_em    @ 13308928    : 256
//   transdist @ 13309184    : 66049 x 256   (row-normalized log transition)
// ---------------------------------------------------------------------------

typedef __attribute__((ext_vector_type(2))) float v2f;
typedef __attribute__((ext_vector_type(8))) float v8f;

#define NK   256
#define ND   256
#define NV   50000
#define NT   512
#define NB   256
#define KP1  257
#define MD   512
#define NROWS (KP1 * KP1)          // 66049
#define VTILE 512
#define NVB   ((NV + VTILE - 1) / VTILE)   // 98

__device__ __forceinline__ v8f wmma_f32(v2f a, v2f b, v8f c) {
  // D = A(16x4, f32) x B(4x16, f32) + C(16x16, f32)
  return __builtin_amdgcn_wmma_f32_16x16x4_f32(
      /*neg_a=*/false, a, /*neg_b=*/false, b,
      /*c_mod=*/(short)0, c, /*reuse_a=*/false, /*reuse_b=*/false);
}

// ---------------- generic tiled transpose: dst[c][r] = src[r][c] ------------
__global__ __launch_bounds__(256)
void transpose_kernel(const float* __restrict__ src, float* __restrict__ dst,
                      int rows, int cols) {
  __shared__ float tile[32][33];
  const int c0 = blockIdx.x * 32, r0 = blockIdx.y * 32;
  const int tx = threadIdx.x & 31, ty = threadIdx.x >> 5;  // 32x8
  for (int i = ty; i < 32; i += 8) {
    int r = r0 + i, c = c0 + tx;
    tile[i][tx] = (r < rows && c < cols) ? src[(size_t)r * cols + c] : 0.f;
  }
  __syncthreads();
  for (int i = ty; i < 32; i += 8) {
    int dr = c0 + i;   // dst row (= src col)
    int dc = r0 + tx;  // dst col (= src row)
    if (dr < cols && dc < rows) dst[(size_t)dr * rows + dc] = tile[tx][i];
  }
}

// ------- emission hidden: h = LN(lembs + relu(lembs @ em_W^T + b)) ----------
// grid = 16 blocks (16 rows each), block = 512 (16 waves); wave w -> col tile w
__global__ __launch_bounds__(512)
void em_hidden_kernel(const float* __restrict__ lembs, const float* __restrict__ em_W,
                      const float* __restrict__ em_bias, const float* __restrict__ em_g,
                      const float* __restrict__ em_beta, float* __restrict__ h_out) {
  __shared__ float aT[16 * ND];
  __shared__ float hT[16 * ND];
  const int r0 = blockIdx.x * 16;
  const int tid = threadIdx.x, lane = tid & 31, wv = tid >> 5;
  for (int e = tid; e < 16 * ND; e += 512)
    aT[e] = lembs[(r0 + (e >> 8)) * ND + (e & 255)];
  __syncthreads();

  const int mh = lane >> 4, ml = lane & 15;
  const int c0 = wv * 16, n = c0 + ml;
  v8f acc = {};
  for (int kk = 0; kk < ND; kk += 4) {
    v2f a, b;
    a.x = aT[ml * ND + kk + 2 * mh];
    a.y = aT[ml * ND + kk + 2 * mh + 1];
    b.x = em_W[n * ND + kk + 2 * mh];      // B[k][n] = em_W[n][k]
    b.y = em_W[n * ND + kk + 2 * mh + 1];
    acc = wmma_f32(a, b, acc);
  }
  const float bias = em_bias[n];
  for (int r = 0; r < 8; ++r) {
    int m = r + 8 * mh;
    float v = acc[r] + bias;
    v = v > 0.f ? v : 0.f;
    hT[m * ND + n] = aT[m * ND + n] + v;   // residual
  }
  __syncthreads();

  // layernorm: wave w owns row w
  const int m = wv;
  float s = 0.f, sq = 0.f;
  for (int e = lane; e < ND; e += 32) { float v = hT[m * ND + e]; s += v; sq += v * v; }
  for (int off = 16; off > 0; off >>= 1) { s += __shfl_xor(s, off, 32); sq += __shfl_xor(sq, off, 32); }
  float mu = s / (float)ND;
  float rstd = rsqrtf(sq / (float)ND - mu * mu + 1e-5f);
  for (int e = lane; e < ND; e += 32)
    h_out[(r0 + m) * ND + e] = (hT[m * ND + e] - mu) * rstd * em_g[e] + em_beta[e];
}

// -------- emission logsumexp partials over vocab tiles ----------------------
// grid = (NVB, K/16); block = 256 (8 waves); each wave does 4 16x16 subtiles
__global__ __launch_bounds__(256)
void em_lse_partial_kernel(const float* __restrict__ h, const float* __restrict__ decWT,
                           const float* __restrict__ dec_b, float* __restrict__ part) {
  __shared__ float hT[16 * ND];     // 16 KB
  __shared__ float lg[16 * VTILE];  // 32 KB
  const int s0 = blockIdx.y * 16;
  const int v0 = blockIdx.x * VTILE;
  const int tid = threadIdx.x, lane = tid & 31, wv = tid >> 5;
  for (int e = tid; e < 16 * ND; e += 256)
    hT[e] = h[(s0 + (e >> 8)) * ND + (e & 255)];
  __syncthreads();

  const int mh = lane >> 4, ml = lane & 15;
  for (int q = 0; q < 4; ++q) {
    const int c0 = 16 * (wv + 8 * q);
    const int n = v0 + c0 + ml;              // vocab column
    const bool ok = (n < NV);
    v8f acc = {};
    for (int kk = 0; kk < ND; kk += 4) {
      v2f a, b;
      a.x = hT[ml * ND + kk + 2 * mh];
      a.y = hT[ml * ND + kk + 2 * mh + 1];
      b.x = ok ? decWT[(size_t)(kk + 2 * mh) * NV + n] : 0.f;
      b.y = ok ? decWT[(size_t)(kk + 2 * mh + 1) * NV + n] : 0.f;
      acc = wmma_f32(a, b, acc);
    }
    const float bias = ok ? dec_b[n] : 0.f;
    for (int r = 0; r < 8; ++r) {
      int m = r + 8 * mh;
      lg[m * VTILE + c0 + ml] = ok ? (acc[r] + bias) : -__builtin_inff();
    }
  }
  __syncthreads();

  for (int m = wv; m < 16; m += 8) {
    float mx = -__builtin_inff();
    for (int e = lane; e < VTILE; e += 32) mx = fmaxf(mx, lg[m * VTILE + e]);
    for (int off = 16; off > 0; off >>= 1) mx = fmaxf(mx, __shfl_xor(mx, off, 32));
    float se = 0.f;
    for (int e = lane; e < VTILE; e += 32) se += __expf(lg[m * VTILE + e] - mx);
    for (int off = 16; off > 0; off >>= 1) se += __shfl_xor(se, off, 32);
    if (lane == 0) {
      size_t idx = ((size_t)(s0 + m) * gridDim.x + blockIdx.x) * 2;
      part[idx] = mx; part[idx + 1] = se;
    }
  }
}

__global__ __launch_bounds__(256)
void em_lse_reduce_kernel(const float* __restrict__ part, float* __restrict__ lse) {
  const int k = threadIdx.x;
  float M = -__builtin_inff();
  for (int j = 0; j < NVB; ++j) M = fmaxf(M, part[((size_t)k * NVB + j) * 2]);
  float s = 0.f;
  for (int j = 0; j < NVB; ++j)
    s += part[((size_t)k * NVB + j) * 2 + 1] * expf(part[((size_t)k * NVB + j) * 2] - M);
  lse[k] = M + logf(s);
}

// -------- emission gather: out[b] = sum_t h[z]·dec_W[x] + b[x] - lse[z] -----
__global__ __launch_bounds__(256)
void emlps_kernel(const int* __restrict__ x, const int* __restrict__ z,
                  const float* __restrict__ h, const float* __restrict__ dec_W,
                  const float* __restrict__ dec_b, const float* __restrict__ lse,
                  float* __restrict__ out) {
  const int b = blockIdx.x;
  const int tid = threadIdx.x, lane = tid & 31, wv = tid >> 5;
  __shared__ float sh[8];
  float acc = 0.f;
  for (int t = wv; t < NT; t += 8) {
    const int zk = z[t * NB + b];
    const int xv = x[t * NB + b];
    const float* hr = h + zk * ND;
    const float* wr = dec_W + (size_t)xv * ND;
    float d = 0.f;
    for (int e = lane; e < ND; e += 32) d += hr[e] * wr[e];
    for (int off = 16; off > 0; off >>= 1) d += __shfl_xor(d, off, 32);
    if (lane == 0) acc += d + dec_b[xv] - lse[zk];
  }
  if (lane == 0) sh[wv] = acc;
  __syncthreads();
  if (tid == 0) {
    float s = 0.f;
    for (int i = 0; i < 8; ++i) s += sh[i];
    out[b] = s;  // emission part written first; trans gather accumulates later
  }
}

// -------- fused transition: cat -> GEMM1+relu+res -> LN -> GEMM2 -> logsoftmax
// grid = ceil(66049/16); block = 512 (16 waves). 64 KB LDS, aliased phases.
__global__ __launch_bounds__(512)
void trans_kernel(const float* __restrict__ tlembs, const float* __restrict__ tmWT,
                  const float* __restrict__ tm_bias, const float* __restrict__ tn_g,
                  const float* __restrict__ tn_beta, const float* __restrict__ tdWT,
                  const float* __restrict__ td_b, float* __restrict__ transdist) {
  __shared__ float smem[16 * MD * 2];   // 64 KB
  float* catL = smem;                   // 16 x 512
  float* thL  = smem + 16 * MD;         // 16 x 512
  float* lgL  = smem;                   // 16 x 256 (aliases catL; catL dead by then)
  const int r0 = blockIdx.x * 16;
  const int tid = threadIdx.x, lane = tid & 31, wv = tid >> 5;

  for (int e = tid; e < 16 * MD; e += 512) {
    int m = e >> 9, c = e & 511;
    int r = r0 + m; if (r >= NROWS) r = 0;            // pad-safe
    int i = r / KP1, j = r % KP1;
    catL[e] = (c < ND) ? tlembs[i * ND + c] : tlembs[j * ND + (c - ND)];
  }
  __syncthreads();

  const int mh = lane >> 4, ml = lane & 15;
  // GEMM1: th_pre (16 x 512); each wave computes 2 column tiles
  for (int q = 0; q < 2; ++q) {
    const int n = 16 * (wv + 16 * q) + ml;
    v8f acc = {};
    for (int kk = 0; kk < MD; kk += 4) {
      v2f a, b;
      a.x = catL[ml * MD + kk + 2 * mh];
      a.y = catL[ml * MD + kk + 2 * mh + 1];
      b.x = tmWT[(kk + 2 * mh) * MD + n];             // B[k][n] = tm_W[n][k]
      b.y = tmWT[(kk + 2 * mh + 1) * MD + n];
      acc = wmma_f32(a, b, acc);
    }
    const float bias = tm_bias[n];
    for (int r = 0; r < 8; ++r) {
      int m = r + 8 * mh;
      float v = acc[r] + bias;
      v = v > 0.f ? v : 0.f;
      thL[m * MD + n] = catL[m * MD + n] + v;
    }
  }
  __syncthreads();

  // layernorm (width 512): wave w owns row w
  {
    const int m = wv;
    float s = 0.f, sq = 0.f;
    for (int e = lane; e < MD; e += 32) { float v = thL[m * MD + e]; s += v; sq += v * v; }
    for (int off = 16; off > 0; off >>= 1) { s += __shfl_xor(s, off, 32); sq += __shfl_xor(sq, off, 32); }
    float mu = s / (float)MD;
    float rstd = rsqrtf(sq / (float)MD - mu * mu + 1e-5f);
    for (int e = lane; e < MD; e += 32)
      thL[m * MD + e] = (thL[m * MD + e] - mu) * rstd * tn_g[e] + tn_beta[e];
  }
  __syncthreads();

  // GEMM2: logits (16 x 256); wave w -> columns [16w, 16w+16)
  {
    const int n = 16 * wv + ml;
    v8f acc = {};
    for (int kk = 0; kk < MD; kk += 4) {
      v2f a, b;
      a.x = thL[ml * MD + kk + 2 * mh];
      a.y = thL[ml * MD + kk + 2 * mh + 1];
      b.x = tdWT[(kk + 2 * mh) * NK + n];             // B[k][n] = td_W[n][k]
      b.y = tdWT[(kk + 2 * mh + 1) * NK + n];
      acc = wmma_f32(a, b, acc);
    }
    const float bias = td_b[n];
    for (int r = 0; r < 8; ++r)
      lgL[(r + 8 * mh) * NK + n] = acc[r] + bias;
  }
  __syncthreads();

  // row log-softmax + store: wave w owns row w
  {
    const int m = wv, r = r0 + m;
    float mx = -__builtin_inff();
    for (int e = lane; e < NK; e += 32) mx = fmaxf(mx, lgL[m * NK + e]);
    for (int off = 16; off > 0; off >>= 1) mx = fmaxf(mx, __shfl_xor(mx, off, 32));
    float se = 0.f;
    for (int e = lane; e < NK; e += 32) se += __expf(lgL[m * NK + e] - mx);
    for (int off = 16; off > 0; off >>= 1) se += __shfl_xor(se, off, 32);
    const float l = mx + __logf(se);
    if (r < NROWS)
      for (int e = lane; e < NK; e += 32)
        transdist[(size_t)r * NK + e] = lgL[m * NK + e] - l;
  }
}

// -------- transition gather: out[b] += sum_t transdist[z(t-2),z(t-1)][z(t)] -
__global__ __launch_bounds__(256)
void trans_gather_kernel(const int* __restrict__ z, const float* __restrict__ transdist,
                         float* __restrict__ out) {
  const int b = blockIdx.x, tid = threadIdx.x;
  __shared__ float sh[256];
  float acc = 0.f;
  for (int t = tid; t < NT; t += 256) {
    const int zt  = z[t * NB + b];
    const int zm1 = (t >= 1) ? z[(t - 1) * NB + b] : NK;
    const int zm2 = (t >= 2) ? z[(t - 2) * NB + b] : NK;
    acc += transdist[(size_t)(zm2 * KP1 + zm1) * NK + zt];
  }
  sh[tid] = acc;
  __syncthreads();
  for (int s = 128; s > 0; s >>= 1) {
    if (tid < s) sh[tid] += sh[tid + s];
    __syncthreads();
  }
  if (tid == 0) out[b] += sh[0];
}

// ---------------------------------------------------------------------------
extern "C" void kernel_launch(void* const* d_in, const int* in_sizes, int n_in,
                              void* d_out, int out_size, void* d_ws, size_t ws_size,
                              hipStream_t stream) {
  (void)in_sizes; (void)n_in; (void)out_size; (void)ws_size;
  const int*   x       = (const int*)  d_in[0];
  const int*   z       = (const int*)  d_in[1];
  const float* lembs   = (const float*)d_in[2];
  const float* tlembs  = (const float*)d_in[3];
  const float* dec_W   = (const float*)d_in[4];
  const float* dec_b   = (const float*)d_in[5];
  const float* em_W    = (const float*)d_in[6];
  const float* em_bias = (const float*)d_in[7];
  const float* em_g    = (const float*)d_in[8];
  const float* em_beta = (const float*)d_in[9];
  const float* td_W    = (const float*)d_in[10];
  const float* td_b    = (const float*)d_in[11];
  const float* tm_W    = (const float*)d_in[12];
  const float* tm_bias = (const float*)d_in[13];
  const float* tn_g    = (const float*)d_in[14];
  const float* tn_beta = (const float*)d_in[15];
  float* out = (float*)d_out;

  float* ws = (float*)d_ws;
  float* decWT   = ws;                       // 256 x 50000
  float* tmWT    = ws + 12800000;            // 512 x 512
  float* tdWT    = ws + 13062144;            // 512 x 256
  float* h_em    = ws + 13193216;            // 256 x 256
  float* lsePart = ws + 13258752;            // 256 x 98 x 2
  float* lse_em  = ws + 13308928;            // 256
  float* trans   = ws + 13309184;            // 66049 x 256

  // one-time operand transposes (coalesced WMMA B-operand access)
  transpose_kernel<<<dim3((ND + 31) / 32, (NV + 31) / 32), 256, 0, stream>>>(dec_W, decWT, NV, ND);
  transpose_kernel<<<dim3(16, 16), 256, 0, stream>>>(tm_W, tmWT, MD, MD);
  transpose_kernel<<<dim3(16, 8),  256, 0, stream>>>(td_W, tdWT, NK, MD);

  // emission path
  em_hidden_kernel<<<16, 512, 0, stream>>>(lembs, em_W, em_bias, em_g, em_beta, h_em);
  em_lse_partial_kernel<<<dim3(NVB, NK / 16), 256, 0, stream>>>(h_em, decWT, dec_b, lsePart);
  em_lse_reduce_kernel<<<1, 256, 0, stream>>>(lsePart, lse_em);
  emlps_kernel<<<NB, 256, 0, stream>>>(x, z, h_em, dec_W, dec_b, lse_em, out);

  // transition path
  trans_kernel<<<(NROWS + 15) / 16, 512, 0, stream>>>(tlembs, tmWT, tm_bias, tn_g, tn_beta,
                                                      tdWT, td_b, trans);
  trans_gather_kernel<<<NB, 256, 0, stream>>>(z, trans, out);
}